// SmoothSTEMinMax_33758442947089
// MI455X (gfx1250) — compile-verified
//
#include <hip/hip_runtime.h>

// SmoothSTEMinMax forward:
//   out[b,o] = (hard - smooth) + smooth,  hard = min_i max(x[b,i], w[i,o]),
//   smooth   = -(1/A) * log( sum_i exp(-A * max(x[b,i], w[i,o])) ),  A = 5.
// x: [B=4096, K=256] f32, w: [K=256, O=256] f32, out: [B, O] f32.

#define A_PARAM   5.0f
#define K_DIM     256
#define O_DIM     256
#define TB        8            // batch rows per block
#define KT        16           // K rows per LDS tile
#define NT        (K_DIM / KT) // 16 tiles
#define TILE_B    (KT * O_DIM * 4)   // 16384 bytes per tile
#define CHUNKS    (TILE_B / (256 * 16)) // 4 b128 chunks per thread per tile

typedef const void __attribute__((address_space(3)))* lds_cptr_t;

__device__ __forceinline__ unsigned lds_off_u32(const void* p) {
    // generic -> LDS addrspace -> raw 32-bit LDS byte offset
    return (unsigned)(unsigned long long)(lds_cptr_t)p;
}

__device__ __forceinline__ void async_copy_b128(unsigned lds_off, unsigned long long gaddr) {
    // GLOBAL_LOAD_ASYNC_TO_LDS_B128 (gfx1250): VDST = LDS byte address VGPR,
    // VADDR = 64-bit global address VGPR pair, GV mode (saddr = off).
    // Tracked by ASYNCcnt.
    asm volatile("global_load_async_to_lds_b128 %0, %1, off"
                 :: "v"(lds_off), "v"(gaddr)
                 : "memory");
}

__device__ __forceinline__ void wait_async_all() {
#if __has_builtin(__builtin_amdgcn_s_wait_asynccnt)
    __builtin_amdgcn_s_wait_asynccnt(0);
#else
    asm volatile("s_wait_asynccnt 0x0" ::: "memory");
#endif
}

__global__ __launch_bounds__(256) void SmoothSTEMinMax_33758442947089_kernel(
    const float* __restrict__ x,
    const float* __restrict__ w,
    float* __restrict__ out,
    int B)
{
    __shared__ float wt[2][KT][O_DIM];   // double-buffered weight tiles, 32 KB

    const int tid = threadIdx.x;         // output column o
    const int b0  = blockIdx.x * TB;

    const unsigned           lds_w = lds_off_u32(&wt[0][0][0]);
    const unsigned long long gw    = (unsigned long long)w;

    // ---- issue async DMA for tile 0 into buffer 0 ----
#pragma unroll
    for (int j = 0; j < CHUNKS; ++j) {
        const unsigned           loff = lds_w + (unsigned)(j * 4096 + tid * 16);
        const unsigned long long ga   = gw + (unsigned long long)(j * 4096 + tid * 16);
        async_copy_b128(loff, ga);
    }

    float h[TB], s[TB];
#pragma unroll
    for (int r = 0; r < TB; ++r) { h[r] = __builtin_inff(); s[r] = 0.0f; }

    wait_async_all();
    __syncthreads();                     // tile 0 resident in LDS for all waves

    for (int k = 0; k < NT; ++k) {
        const int cur = k & 1;

        // ---- prefetch tile k+1 into the other buffer (overlaps compute) ----
        if (k + 1 < NT) {
            const int nxt = (k + 1) & 1;
            const unsigned           lbase = lds_w + (unsigned)(nxt * TILE_B);
            const unsigned long long gbase = gw + (unsigned long long)(k + 1) * TILE_B;
#pragma unroll
            for (int j = 0; j < CHUNKS; ++j) {
                const unsigned           loff = lbase + (unsigned)(j * 4096 + tid * 16);
                const unsigned long long ga   = gbase + (unsigned long long)(j * 4096 + tid * 16);
                async_copy_b128(loff, ga);
            }
        }

        // ---- register-cache this thread's w column chunk (16 ds_load_b32) ----
        float wreg[KT];
#pragma unroll
        for (int ii = 0; ii < KT; ++ii) wreg[ii] = wt[cur][ii][tid];

        // ---- pure-VALU inner loops; x comes in via uniform scalar loads ----
#pragma unroll
        for (int r = 0; r < TB; ++r) {
            int row = b0 + r;
            if (row >= B) row = B - 1;               // clamp (B divisible by TB anyway)
            const float* xr = x + (size_t)row * K_DIM + k * KT;
            float hh = h[r], ss = s[r];
#pragma unroll
            for (int ii = 0; ii < KT; ++ii) {
                const float m = fmaxf(xr[ii], wreg[ii]);  // v_max_f32 (sgpr, vgpr)
                hh = fminf(hh, m);                        // v_min_f32
                ss += __expf(-A_PARAM * m);               // v_mul + v_exp_f32 + v_add
            }
            h[r] = hh; s[r] = ss;
        }

        if (k + 1 < NT) {
            wait_async_all();            // my waves' tile-(k+1) chunks landed
            __syncthreads();             // all waves' chunks landed; buf free for reuse
        }
    }

    // ---- epilogue: smooth term and straight-through combine, coalesced stores ----
#pragma unroll
    for (int r = 0; r < TB; ++r) {
        const int row = b0 + r;
        if (row < B) {
            const float smooth = -(1.0f / A_PARAM) * __logf(s[r]);
            out[(size_t)row * O_DIM + tid] = (h[r] - smooth) + smooth;
        }
    }
}

extern "C" void kernel_launch(void* const* d_in, const int* in_sizes, int n_in,
                              void* d_out, int out_size, void* d_ws, size_t ws_size,
                              hipStream_t stream) {
    (void)n_in; (void)out_size; (void)d_ws; (void)ws_size;
    const float* x = (const float*)d_in[0];   // [B, 256]
    const float* w = (const float*)d_in[1];   // [256, 256]
    float* out     = (float*)d_out;           // [B, 256]

    const int B = in_sizes[0] / K_DIM;        // 4096
    const int grid = (B + TB - 1) / TB;       // 512 blocks

    SmoothSTEMinMax_33758442947089_kernel<<<grid, 256, 0, stream>>>(x, w, out, B);
}